// FireflyOptimizer_1726576855101
// MI455X (gfx1250) — compile-verified
//
#include <hip/hip_runtime.h>

// ---------------------------------------------------------------------------
// Firefly optimizer on MI455X (gfx1250).
//  - Gram matrix (x x^T) and attraction (W x, W 1) via V_WMMA_F32_16X16X4_F32
//  - f32 matrix cores keep reference (f32) precision for exp(-d2)
//  - per-step kernel chain on the stream provides the global sync
// ---------------------------------------------------------------------------

typedef float v2f __attribute__((ext_vector_type(2)));
typedef float v8f __attribute__((ext_vector_type(8)));

#define SPECIES 4
#define POP     4096
#define DIM     16
#define STEPS   4
#define NSPLIT  4                       // j-range split per i-tile
#define JTILES  (POP / 16)              // 256
#define TILES_PER_SPLIT (JTILES / NSPLIT)
#define LDS_STRIDE 20                   // 16x20 f32 slab: even (b64 align), halves 32 banks apart

#define V8Z {0.f,0.f,0.f,0.f,0.f,0.f,0.f,0.f}

__device__ __forceinline__ float exp_neg(float d2) {
#if __has_builtin(__builtin_amdgcn_exp2f)
  return __builtin_amdgcn_exp2f(-1.44269504088896340736f * d2); // v_exp_f32
#else
  return __expf(-d2);
#endif
}

__device__ __forceinline__ float cos2pi(float x) {
#if __has_builtin(__builtin_amdgcn_cosf)
  return __builtin_amdgcn_cosf(x);      // v_cos_f32 computes cos(2*pi*x)
#else
  return __cosf(6.28318530717958647692f * x);
#endif
}

// ------------------------- kernel A: costs + squared norms ------------------
__global__ void __launch_bounds__(256)
k_cost(const float* __restrict__ x, float* __restrict__ cost, float* __restrict__ sq) {
  const int idx = blockIdx.x * blockDim.x + threadIdx.x;   // 0 .. S*POP-1
  const float* row = x + (size_t)idx * DIM;
  float s2 = 0.f, sc = 0.f;
#pragma unroll
  for (int d = 0; d < DIM; ++d) {
    const float v = row[d];
    s2 += v * v;
    sc += cos2pi(v);
  }
  sq[idx]   = s2;
  cost[idx] = 10.0f * DIM + s2 - 10.0f * sc;
}

// ------------------------- kernel B: pairwise WMMA --------------------------
// wave w -> (species s, i-tile, j-split). Per 16x16 j-tile:
//   4 WMMA : G = x_i x_j^T           (K = dim, chunks of 4)
//   VALU   : d2, mask, W = 2*exp(-d2)
//   LDS    : W tile roundtrip (C-layout -> A-layout)
//   8 WMMA : att += W x_j ; wsum += W 1
__global__ void __launch_bounds__(256)
k_pair(const float* __restrict__ x, const float* __restrict__ cost,
       const float* __restrict__ sq, float* __restrict__ part) {
  __shared__ __attribute__((aligned(16))) float lds[8][16 * LDS_STRIDE];

  const int wave = threadIdx.x >> 5;
  const int lane = threadIdx.x & 31;
  const int col  = lane & 15;
  const int hi   = lane >> 4;

  const int w     = blockIdx.x * 8 + wave;   // 0 .. 4095
  const int s     = w >> 10;                 // 1024 waves / species
  const int rest  = w & 1023;
  const int itile = rest >> 2;
  const int split = rest & 3;
  const int ibase = itile * 16;

  const float* xs = x    + (size_t)s * POP * DIM;
  const float* cs = cost + (size_t)s * POP;
  const float* ss = sq   + (size_t)s * POP;
  float* lw = lds[wave];

  // A operands (x_i rows, reused for the whole j loop)
  v2f A[4];
#pragma unroll
  for (int c = 0; c < 4; ++c)
    A[c] = *(const v2f*)(xs + (size_t)(ibase + col) * DIM + c * 4 + 2 * hi);

  // per-lane row constants in C/D layout (M = r + 8*hi)
  float sqi[8], ci[8], xic[8];
#pragma unroll
  for (int r = 0; r < 8; ++r) {
    const int row = ibase + r + 8 * hi;
    sqi[r] = ss[row];
    ci[r]  = cs[row];
    xic[r] = xs[(size_t)row * DIM + col];
  }

  v8f att  = V8Z;
  v8f wsum = V8Z;
  const v2f ones = {1.0f, 1.0f};

  const int jt0 = split * TILES_PER_SPLIT;
  for (int jt = jt0; jt < jt0 + TILES_PER_SPLIT; ++jt) {
    const int jb = jt * 16;

    // ---- Gram tile: G[i,j] = sum_d x_i[i,d] * x_j[j,d]
    v8f g = V8Z;
#pragma unroll
    for (int c = 0; c < 4; ++c) {
      const v2f B = *(const v2f*)(xs + (size_t)(jb + col) * DIM + c * 4 + 2 * hi);
      g = __builtin_amdgcn_wmma_f32_16x16x4_f32(false, A[c], false, B,
                                                (short)0, g, false, false);
    }

    const float sqj = ss[jb + col];
    const float cj  = cs[jb + col];

    // ---- W = mask * 2*exp(-d2), stored to LDS in row-major 16x20
#pragma unroll
    for (int r = 0; r < 8; ++r) {
      const float d2 = fmaxf(sqi[r] + sqj - 2.0f * g[r], 0.0f);
      const float wv = (ci[r] > cj) ? (2.0f * exp_neg(d2)) : 0.0f;
      lw[(r + 8 * hi) * LDS_STRIDE + col] = wv;
    }
    asm volatile("s_wait_dscnt 0" ::: "memory");

    // ---- att += W * x_j ; wsum += W * ones   (A = W from LDS, K over j)
#pragma unroll
    for (int c = 0; c < 4; ++c) {
      const v2f WA = *(const v2f*)(lw + col * LDS_STRIDE + c * 4 + 2 * hi);
      const int jr = jb + c * 4 + 2 * hi;
      const v2f BX = { xs[(size_t)jr * DIM + col],
                       xs[(size_t)(jr + 1) * DIM + col] };
      att  = __builtin_amdgcn_wmma_f32_16x16x4_f32(false, WA, false, BX,
                                                   (short)0, att, false, false);
      wsum = __builtin_amdgcn_wmma_f32_16x16x4_f32(false, WA, false, ones,
                                                   (short)0, wsum, false, false);
    }
  }

  // attraction partial = (W@x)[i,d] - Wsum[i] * x[i,d]
  float* po = part + ((size_t)split * SPECIES + s) * POP * DIM;
#pragma unroll
  for (int r = 0; r < 8; ++r) {
    const int row = ibase + r + 8 * hi;
    po[(size_t)row * DIM + col] = att[r] - wsum[r] * xic[r];
  }
}

// ------------------------- kernel C: update + clip + mixing -----------------
__global__ void __launch_bounds__(256)
k_update(const float* __restrict__ xcur, const float* __restrict__ part,
         const float* __restrict__ noise, float* __restrict__ xnext,
         float alpha, int mix) {
  const int idx = blockIdx.x * blockDim.x + threadIdx.x;  // row id: s*POP + p
  const int s = idx / POP;
  const int p = idx - s * POP;
  const float* xr = xcur  + (size_t)idx * DIM;
  const float* nr = noise + (size_t)idx * DIM;

  int ds_ = s;
  if (mix && p >= (POP / 2)) ds_ = (s + 1) & 3;           // roll(+1) fold at write
  float* orow = xnext + ((size_t)ds_ * POP + p) * DIM;

#pragma unroll
  for (int d = 0; d < DIM; ++d) {
    float acc = xr[d];
#pragma unroll
    for (int k = 0; k < NSPLIT; ++k)
      acc += part[((size_t)k * SPECIES * POP + idx) * DIM + d];
    acc += alpha * (nr[d] - 0.5f) * 8.0f;                 // (UB-LB) = 8
    orow[d] = fminf(fmaxf(acc, -4.0f), 4.0f);
  }
}

// ---------------------------------------------------------------------------
extern "C" void kernel_launch(void* const* d_in, const int* in_sizes, int n_in,
                              void* d_out, int out_size, void* d_ws, size_t ws_size,
                              hipStream_t stream) {
  (void)in_sizes; (void)n_in; (void)out_size; (void)ws_size;
  const float* fireflies = (const float*)d_in[0];          // [4,4096,16]
  const float* noise     = (const float*)d_in[1];          // [4,4,4096,16]
  float* out = (float*)d_out;

  float* ws = (float*)d_ws;
  const size_t NX = (size_t)SPECIES * POP * DIM;           // 262144
  float* xA   = ws;
  float* xB   = xA + NX;
  float* cost = xB + NX;
  float* sq   = cost + (size_t)SPECIES * POP;
  float* part = sq   + (size_t)SPECIES * POP;              // NSPLIT * NX floats

  const float* xcur[STEPS] = { fireflies, xA, xB, xA };
  float*       xnxt[STEPS] = { xA,        xB, xA, out };

  const dim3 blk(256);
  const dim3 gridA((SPECIES * POP) / 256);                 // 64
  const dim3 gridB((SPECIES * JTILES * NSPLIT) / 8);       // 512 (8 waves/block)
  const dim3 gridC((SPECIES * POP) / 256);                 // 64

  double alpha = 0.1;
  for (int step = 0; step < STEPS; ++step) {
    k_cost<<<gridA, blk, 0, stream>>>(xcur[step], cost, sq);
    k_pair<<<gridB, blk, 0, stream>>>(xcur[step], cost, sq, part);
    k_update<<<gridC, blk, 0, stream>>>(xcur[step], part,
                                        noise + (size_t)step * NX,
                                        xnxt[step], (float)alpha,
                                        (step % 25 == 0) ? 1 : 0);
    alpha *= 0.995;
  }
}